// GINLayer_48507360641133
// MI455X (gfx1250) — compile-verified
//
#include <hip/hip_runtime.h>

#define D_FEAT 64
#define EPH 4  // edges per half-wave per iteration

// Native non-returning fp32 atomic add (global_atomic_add_f32).
// Plain atomicAdd(float*) may lower to a CAS loop; unsafeAtomicAdd is the
// HIP path that forces the hardware instruction (fine here: inputs are
// normal-range uniforms/normals, denorm flush semantics irrelevant).
__device__ __forceinline__ void atomic_add_f32(float* p, float v) {
#if defined(__HIP_DEVICE_COMPILE__)
    unsafeAtomicAdd(p, v);
#else
    atomicAdd(p, v);
#endif
}

// ---------------------------------------------------------------------------
// Kernel 1: out = (1 + eps) * x   (float4-vectorized, fully overwrites d_out)
// ---------------------------------------------------------------------------
__global__ __launch_bounds__(256) void gin_init(const float* __restrict__ x,
                                                const float* __restrict__ eps,
                                                float* __restrict__ out,
                                                int n4) {
    int i = blockIdx.x * blockDim.x + threadIdx.x;
    if (i >= n4) return;
    const float s = 1.0f + eps[0];               // uniform -> s_load + s_add_f32
    float4 v = reinterpret_cast<const float4*>(x)[i];
    v.x *= s; v.y *= s; v.z *= s; v.w *= s;
    reinterpret_cast<float4*>(out)[i] = v;
}

// ---------------------------------------------------------------------------
// Kernel 2: scatter-add of w * x[col] into out[row].
// Half-wave per edge: lanes 0-15 of a wave own edge A, lanes 16-31 edge B.
// Each lane owns 4 consecutive features -> one global_load_b128 instruction
// gathers two edges' 64 floats; 4 global_atomic_add_f32 per lane scatter them.
// Self-loops / tail duplicates are masked by wt = 0 (branchless: adds 0.0f).
// ---------------------------------------------------------------------------
__global__ __launch_bounds__(256) void gin_edges(const float* __restrict__ x,
                                                 const int*   __restrict__ rows,
                                                 const int*   __restrict__ cols,
                                                 const float* __restrict__ w,
                                                 float* __restrict__ out,
                                                 int n_edges) {
    const int fl     = (threadIdx.x & 15) << 2;                     // feature base: 0,4,...,60
    const int half   = (blockIdx.x * blockDim.x + threadIdx.x) >> 4; // global half-wave id
    const int n_half = (gridDim.x * blockDim.x) >> 4;

    for (int base = half * EPH; base < n_edges; base += n_half * EPH) {
        // Prefetch next group's edge stream (global_prefetch_b8).
        const int pf = base + n_half * EPH;
        if (pf < n_edges && (threadIdx.x & 15) == 0) {
            __builtin_prefetch(rows + pf, 0, 0);
            __builtin_prefetch(cols + pf, 0, 0);
            __builtin_prefetch(w + pf, 0, 0);
        }

        int    r[EPH];
        float  wt[EPH];
        float4 xv[EPH];

        // Load edge scalars (streamed once -> non-temporal) and issue all
        // gathers up front for memory-level parallelism. x is L2-resident
        // (12.8 MB of 192 MB), so gathers are L2 hits.
#pragma unroll
        for (int k = 0; k < EPH; ++k) {
            int e = base + k;
            const bool ok = (e < n_edges);
            e = ok ? e : (n_edges - 1);                  // tail clamp, masked below
            r[k] = __builtin_nontemporal_load(rows + e);
            const int   c  = __builtin_nontemporal_load(cols + e);
            const float ww = __builtin_nontemporal_load(w + e);
            wt[k] = (ok && (r[k] != c)) ? ww : 0.0f;     // remove_self_loops
            xv[k] = *reinterpret_cast<const float4*>(x + (size_t)c * D_FEAT + fl);
        }

        // Scatter: branchless non-returning f32 atomics at the L2 atomic units.
#pragma unroll
        for (int k = 0; k < EPH; ++k) {
            float* dst = out + (size_t)r[k] * D_FEAT + fl;
            atomic_add_f32(dst + 0, wt[k] * xv[k].x);
            atomic_add_f32(dst + 1, wt[k] * xv[k].y);
            atomic_add_f32(dst + 2, wt[k] * xv[k].z);
            atomic_add_f32(dst + 3, wt[k] * xv[k].w);
        }
    }
}

// ---------------------------------------------------------------------------
// Launch
// ---------------------------------------------------------------------------
extern "C" void kernel_launch(void* const* d_in, const int* in_sizes, int n_in,
                              void* d_out, int out_size, void* d_ws, size_t ws_size,
                              hipStream_t stream) {
    const float* x          = (const float*)d_in[0];     // [N, 64] f32
    const int*   edge_index = (const int*)  d_in[1];     // [2, E] int32 (row-major)
    const float* edge_w     = (const float*)d_in[2];     // [E] f32
    const float* eps        = (const float*)d_in[3];     // [1] f32
    float*       out        = (float*)d_out;             // [N, 64] f32

    const int n_edges = in_sizes[2];                     // E from edge_weight
    const int n_elems = in_sizes[0];                     // N * 64
    const int* rows = edge_index;                        // edge_index[0]
    const int* cols = edge_index + n_edges;              // edge_index[1]

    // Pass 1: out = (1 + eps) * x
    const int n4 = n_elems / 4;
    gin_init<<<(n4 + 255) / 256, 256, 0, stream>>>(x, eps, out, n4);

    // Pass 2: edge scatter-add (stream-ordered after init).
    // 2048 blocks * 16 half-waves = 32768 half-waves, ~6 groups of 4 edges.
    gin_edges<<<2048, 256, 0, stream>>>(x, rows, cols, edge_w, out, n_edges);
}